// InstanceLoss_4028679324090
// MI455X (gfx1250) — compile-verified
//
#include <hip/hip_runtime.h>
#include <hip/hip_bf16.h>
#include <math.h>

// Problem constants (fixed by the reference)
#define NB   2
#define NA   49104      // total anchors
#define NCLS 80
#define NANN 8
#define HH   512
#define WW   512
#define GBLK 192        // ceil(NA/256); 192*64-aligned for WMMA tiles
#define NQ   27         // partial quantities per image
#define NPAD 32768      // pow2 >= max region (181*181=32761)

// workspace float layout
#define OFF_PART 0
#define PART_SZ  (NB*NQ*GBLK)
#define OFF_CLS  (OFF_PART + PART_SZ)
#define OFF_REG  (OFF_CLS + NB)
#define OFF_S    (OFF_REG + NB)
#define OFF_SIG  (OFF_S   + NB*NANN)
#define OFF_VAL  (OFF_SIG + NB*NANN)
#define OFF_AE   (OFF_VAL + NB*NANN)

typedef __attribute__((ext_vector_type(2))) float v2f;
typedef __attribute__((ext_vector_type(8))) float v8f;

// ---------------------------------------------------------------------------
// Pass A: per-anchor IoU / focal / smooth-L1 / per-annotation sigma partials.
// Each block writes 27 partial sums (SoA) -> deterministic reduction later.
// ---------------------------------------------------------------------------
__global__ __launch_bounds__(256)
void pass_a(const float* __restrict__ cls,
            const float* __restrict__ reg,
            const float* __restrict__ anc,
            const float* __restrict__ bbox,
            float* __restrict__ ws)
{
  int tid = threadIdx.x;
  int img = blockIdx.x / GBLK;
  int blk = blockIdx.x % GBLK;
  int a   = blk * 256 + tid;

  float clsL = 0.f, regL = 0.f, posF = 0.f, rv = 0.f, rv2 = 0.f;
  int annIdx = -1;

  if (a < NA) {
    float a0 = anc[a*4+0], a1 = anc[a*4+1], a2 = anc[a*4+2], a3 = anc[a*4+3];
    float ah = a2 - a0, aw = a3 - a1;
    float acx = a1 + 0.5f*aw, acy = a0 + 0.5f*ah;
    float aarea = ah * aw;

    float best = -1.f; int barg = 0;
    for (int n = 0; n < NANN; n++) {
      const float* b = bbox + ((img*NANN) + n) * 5;
      float bx1 = b[0], by1 = b[1], bx2 = b[2], by2 = b[3];
      float area = (bx2 - bx1) * (by2 - by1);
      float iw = fmaxf(fminf(a3, bx2) - fmaxf(a1, bx1), 0.f);
      float ih = fmaxf(fminf(a2, by2) - fmaxf(a0, by1), 0.f);
      float ua = fmaxf(aarea + area - iw*ih, 1e-8f);
      float iou = iw * ih / ua;
      if (iou > best) { best = iou; barg = n; }   // first-max like jnp.argmax
    }
    bool pos = best >= 0.5f;
    bool neg = best <  0.4f;
    const float* asg = bbox + ((img*NANN) + barg) * 5;
    int acls = (int)asg[4];

    const float* prow = cls + ((size_t)(img*NA) + a) * NCLS;
    __builtin_prefetch(prow, 0, 0);   // global_prefetch_b8
    if (pos || neg) {
      for (int c = 0; c < NCLS; c++) {
        float p = fminf(fmaxf(prow[c], 1e-4f), 1.f - 1e-4f);
        bool t1 = pos && (c == acls);
        float fw, bce;
        if (t1) { float q = 1.f - p; fw = 0.25f * q * q; bce = -logf(p); }
        else    { fw = 0.75f * p * p; bce = -logf(1.f - p); }
        clsL += fw * bce;
      }
    }
    const float* rrow = reg + ((size_t)(img*NA) + a) * 5;
    if (pos) {
      posF = 1.f; annIdx = barg;
      float gw0 = asg[2]-asg[0], gh0 = asg[3]-asg[1];
      float gcx = asg[0] + 0.5f*gw0, gcy = asg[1] + 0.5f*gh0;
      float gw = fmaxf(gw0, 1.f), gh = fmaxf(gh0, 1.f);
      float tt[4] = { (gcy-acy)/ah, (gcx-acx)/aw, logf(gh/ah), logf(gw/aw) };
      for (int k = 0; k < 4; k++) {
        float d = fabsf(tt[k] - rrow[k]);
        regL += (d <= 1.f/9.f) ? 4.5f*d*d : d - 0.5f/9.f;
      }
      rv = rrow[4]; rv2 = rv * rv;
    }
  }

  __shared__ float sdata[256];
  for (int q = 0; q < NQ; q++) {
    float v;
    if      (q == 0) v = clsL;
    else if (q == 1) v = regL;
    else if (q == 2) v = posF;
    else if (q < 11) v = (annIdx == q-3)  ? 1.f : 0.f;
    else if (q < 19) v = (annIdx == q-11) ? rv  : 0.f;
    else             v = (annIdx == q-19) ? rv2 : 0.f;
    sdata[tid] = v; __syncthreads();
    for (int s2 = 128; s2 > 0; s2 >>= 1) {
      if (tid < s2) sdata[tid] += sdata[tid + s2];
      __syncthreads();
    }
    if (tid == 0) ws[OFF_PART + ((img*NQ) + q)*GBLK + blk] = sdata[0];
    __syncthreads();
  }
}

// ---------------------------------------------------------------------------
// Finalize 1: reduce 192 partials per quantity with V_WMMA_F32_16X16X4_F32.
// A = ones(16x4) -> D rows = column sums of B; chain C over 3 tiles of 64.
// One wave (EXEC all ones, as WMMA requires).
// ---------------------------------------------------------------------------
__global__ __launch_bounds__(32)
void finalize1(float* __restrict__ ws)
{
  int lane = threadIdx.x;
  __shared__ float sums[NQ];
  __shared__ float lbuf[32];
  v2f ones; ones.x = 1.f; ones.y = 1.f;

  for (int img = 0; img < NB; img++) {
    for (int q = 0; q < NQ; q++) {
      const float* P = ws + OFF_PART + ((img*NQ) + q) * GBLK;
      v8f c = {0.f,0.f,0.f,0.f,0.f,0.f,0.f,0.f};
      for (int t = 0; t < 3; t++) {
        v2f b; b.x = P[t*64 + 2*lane]; b.y = P[t*64 + 2*lane + 1];
        c = __builtin_amdgcn_wmma_f32_16x16x4_f32(
                false, ones, false, b, (short)0, c, false, false);
      }
      lbuf[lane] = c[0];            // D row M=0 lives in VGPR0 lanes 0..15
      __syncthreads();
      if (lane == 0) { float s = 0.f; for (int l = 0; l < 16; l++) s += lbuf[l]; sums[q] = s; }
      __syncthreads();
    }
    if (lane == 0) {
      float npos = sums[2];
      float nposf = fmaxf(npos, 1.f);
      ws[OFF_CLS + img] = sums[0] / nposf;
      ws[OFF_REG + img] = (npos > 0.f) ? sums[1] / fmaxf(4.f*npos, 1.f) : 0.f;
      for (int n = 0; n < NANN; n++) {
        float cnt = sums[3+n], sr = sums[11+n], sr2 = sums[19+n];
        float cntf = fmaxf(cnt, 1.f);
        float ts = sr / cntf;
        float sig = (sr2 - 2.f*ts*sr + cnt*ts*ts) / cntf;
        ws[OFF_S   + img*NANN + n] = expf(ts);
        ws[OFF_SIG + img*NANN + n] = sig;
        ws[OFF_VAL + img*NANN + n] = (cnt > 0.f) ? 1.f : 0.f;
      }
    }
    __syncthreads();
  }
}

// ---------------------------------------------------------------------------
// Pass B: Lovasz hinge per (image, annotation). Sort region pixels only
// (out-of-region entries are provably inert), bitonic sort of packed
// (error,label) keys in dynamic LDS, then scan + gradient sum.
// ---------------------------------------------------------------------------
__global__ __launch_bounds__(1024)
void pass_b(const float* __restrict__ assoc,   // (B,2,H,W), reference uses image 0 only
            const float* __restrict__ bbox,
            const int*   __restrict__ ids,
            const int*   __restrict__ imap,
            const float* __restrict__ xym,     // (2,1024,1024)
            float* __restrict__ ws)
{
  extern __shared__ char smem[];
  unsigned* keys = (unsigned*)smem;                 // NPAD u32 = 128 KB
  float*    tsum = (float*)(smem + NPAD * 4);       // 1024 floats
  __shared__ float sp[4];
  __shared__ int   ip[8];

  int tid = threadIdx.x;
  int img = blockIdx.x / NANN;
  int ann = blockIdx.x % NANN;

  if (tid == 0) {
    const float* ac = bbox + (img*NANN + ann) * 5;
    int olt0 = (int)ac[1], olt1 = (int)ac[0];       // [y1,x1]
    int orb0 = (int)ac[3], orb1 = (int)ac[2];       // [y2,x2]
    int cen0 = (int)((float)(olt0 + orb0) * 0.5f);
    int cen1 = (int)((float)(olt1 + orb1) * 0.5f);
    float ctr0 = (float)(olt0 + orb0) * 0.5f, ctr1 = (float)(olt1 + orb1) * 0.5f;
    float hf0 = (float)(orb0 - olt0) * 0.75f, hf1 = (float)(orb1 - olt1) * 0.75f;
    int lt0 = (int)fmaxf(floorf(ctr0 - hf0), 0.f);
    int lt1 = (int)fmaxf(floorf(ctr1 - hf1), 0.f);
    int rb0 = (int)fminf(ceilf(ctr0 + hf0), (float)HH);
    int rb1 = (int)fminf(ceilf(ctr1 + hf1), (float)WW);
    int rw = rb1 - lt1; if (rw < 0) rw = 0;
    int rh = rb0 - lt0; if (rh < 0) rh = 0;
    int cnt = rw * rh; if (cnt > NPAD) cnt = NPAD;
    ip[0] = lt0; ip[1] = lt1; ip[2] = rw; ip[3] = cnt;
    ip[4] = ids[img*NANN + ann];
    sp[0] = ws[OFF_S + img*NANN + ann];             // s = exp(ts)
    sp[1] = xym[cen0*1024 + cen1];                  // center ch0
    sp[2] = xym[1024*1024 + cen0*1024 + cen1];      // center ch1
  }
  __syncthreads();
  int lt0 = ip[0], lt1 = ip[1], rw = ip[2], count = ip[3], iid = ip[4];
  float s = sp[0], c0 = sp[1], c1 = sp[2];

  // build packed keys: sortable(err) with label in the LSB
  for (int i = tid; i < NPAD; i += 1024) {
    unsigned key = 0u;                              // padding sorts last
    if (i < count) {
      int ry = i / rw, rx = i - ry*rw;
      int gy = lt0 + ry, gx = lt1 + rx;
      float a0v = assoc[(size_t)0*HH*WW + (size_t)gy*WW + gx];
      float a1v = assoc[(size_t)1*HH*WW + (size_t)gy*WW + gx];
      float se0 = tanhf(a0v) + xym[gy*1024 + gx];
      float se1 = tanhf(a1v) + xym[1024*1024 + gy*1024 + gx];
      float d0 = se0 - c0, d1 = se1 - c1;
      float dist = expf(-(d0*d0 + d1*d1) * s);
      int lab = (imap[(size_t)img*HH*WW + (size_t)gy*WW + gx] == iid) ? 1 : 0;
      float sign = lab ? 1.f : -1.f;
      float e = 1.f - (2.f*dist - 1.f)*sign;        // e in [0,2]
      unsigned so = __float_as_uint(e) | 0x80000000u;
      key = (so & ~1u) | (unsigned)lab;
    }
    keys[i] = key;
  }
  __syncthreads();

  // bitonic sort, descending
  for (unsigned k = 2; k <= NPAD; k <<= 1) {
    for (unsigned j = k >> 1; j > 0; j >>= 1) {
      for (unsigned i = tid; i < NPAD; i += 1024) {
        unsigned ixj = i ^ j;
        if (ixj > i) {
          unsigned ki = keys[i], kj = keys[ixj];
          bool desc = ((i & k) == 0);
          if (desc ? (ki < kj) : (ki > kj)) { keys[i] = kj; keys[ixj] = ki; }
        }
      }
      __syncthreads();
    }
  }

  // scan labels: per-thread chunk of 32, serial offset scan by t0 (tiny)
  int base = tid * 32;
  float gs = 0.f;
  for (int i = base; i < base + 32; i++) gs += (float)(keys[i] & 1u);
  tsum[tid] = gs;
  __syncthreads();
  if (tid == 0) {
    float run = 0.f;
    for (int t = 0; t < 1024; t++) { float v = tsum[t]; tsum[t] = run; run += v; }
    sp[3] = run;                                    // gts
  }
  __syncthreads();
  float gts = sp[3];
  float cum = tsum[tid];                            // exclusive prefix at base
  float part = 0.f;
  if (base < count) {
    float jprev;
    if (base == 0) jprev = 0.f;
    else {
      float uni = fmaxf(gts + (float)base - cum, 1e-8f);
      jprev = 1.f - (gts - cum) / uni;
    }
    int end = min(base + 32, count);
    for (int i = base; i < end; i++) {
      unsigned key = keys[i];
      float g = (float)(key & 1u);
      float e = __uint_as_float(key & 0x7FFFFFFEu); // e >= 0, so relu(e)=e
      cum += g;
      float uni = fmaxf(gts + (float)(i + 1) - cum, 1e-8f);
      float jc = 1.f - (gts - cum) / uni;
      part += e * (jc - jprev);
      jprev = jc;
    }
  }
  __syncthreads();
  tsum[tid] = part;
  __syncthreads();
  for (int s2 = 512; s2 > 0; s2 >>= 1) {
    if (tid < s2) tsum[tid] += tsum[tid + s2];
    __syncthreads();
  }
  if (tid == 0) {
    float valid = ws[OFF_VAL + img*NANN + ann];
    ws[OFF_AE + img*NANN + ann] = valid * tsum[0];
  }
}

// ---------------------------------------------------------------------------
// Finalize 2: combine into the 4 outputs.
// ---------------------------------------------------------------------------
__global__ void finalize2(const float* __restrict__ ws, float* __restrict__ out)
{
  if (threadIdx.x == 0 && blockIdx.x == 0) {
    float clsm = 0.f, regm = 0.f, sigm = 0.f, aem = 0.f;
    for (int img = 0; img < NB; img++) {
      float nv = 0.f, ssum = 0.f, asum = 0.f;
      for (int n = 0; n < NANN; n++) {
        float v = ws[OFF_VAL + img*NANN + n];
        nv   += v;
        ssum += v * ws[OFF_SIG + img*NANN + n];
        asum += ws[OFF_AE + img*NANN + n];
      }
      float nuni = fmaxf(nv, 1.f);
      clsm += ws[OFF_CLS + img];
      regm += ws[OFF_REG + img];
      sigm += ssum / nuni;
      aem  += asum / nuni;
    }
    out[0] = clsm * 0.5f;
    out[1] = regm * 0.5f * 50.f;
    out[2] = sigm * 0.5f;
    out[3] = aem  * 0.5f;
  }
}

extern "C" void kernel_launch(void* const* d_in, const int* in_sizes, int n_in,
                              void* d_out, int out_size, void* d_ws, size_t ws_size,
                              hipStream_t stream)
{
  const float* cls   = (const float*)d_in[0];
  const float* reg   = (const float*)d_in[1];
  const float* anc   = (const float*)d_in[2];
  const float* assoc = (const float*)d_in[3];
  const float* bbox  = (const float*)d_in[4];
  const int*   ids   = (const int*)d_in[5];
  const int*   imap  = (const int*)d_in[6];
  const float* xym   = (const float*)d_in[7];
  float* ws  = (float*)d_ws;
  float* out = (float*)d_out;

  pass_a<<<NB*GBLK, 256, 0, stream>>>(cls, reg, anc, bbox, ws);
  finalize1<<<1, 32, 0, stream>>>(ws);
  size_t smem = (size_t)NPAD * sizeof(unsigned) + 1024 * sizeof(float);
  pass_b<<<NB*NANN, 1024, smem, stream>>>(assoc, bbox, ids, imap, xym, ws);
  finalize2<<<1, 1, 0, stream>>>(ws, out);
}